// hyperRNN_58952721105618
// MI455X (gfx1250) — compile-verified
//
#include <hip/hip_runtime.h>
#include <math.h>

// ---------------------------------------------------------------------------
// Hyperbolic (Poincare) RNN for MI455X / gfx1250.
// GEMMs run on V_WMMA_F32_16X16X4_F32 (native fp32 tensor op, full precision
// for the 256-step recurrent chain). Ux is hoisted into one large GEMM
// (MT=2: two M-tiles share each B fragment -> half the L2 traffic); Wh is a
// per-step GEMM (MT=1 for max workgroup parallelism) + elementwise combine
// with double-buffered row-norm accumulators.
// ---------------------------------------------------------------------------

typedef float v2f __attribute__((ext_vector_type(2)));
typedef float v8f __attribute__((ext_vector_type(8)));

#define BB   256
#define TT   256
#define HH   512
#define KD   1024          // 2*I
#define ND   1024          // 2*H
#define CHUNK 64           // K-chunk staged in LDS
#define EPSF 1e-15f
#define MAXN (1.0f - 1e-5f)

// ---------------------------------------------------------------------------
// init: zero h state, uxsum accumulators; seed hsum[0] with ||h0+EPS||^2
__global__ __launch_bounds__(256)
void rnn_init_kernel(float* __restrict__ h, float* __restrict__ msum,
                     float* __restrict__ hsum, float* __restrict__ uxsum) {
  int i = blockIdx.x * 256 + threadIdx.x;        // grid covers BB*KD = 262144
  h[i] = 0.0f;
  if (i < BB * TT) uxsum[i] = 0.0f;
  if (i < 2 * BB) {
    msum[i] = 0.0f;
    hsum[i] = (i < BB) ? (float)KD * EPSF * EPSF : 0.0f;   // ||0+EPS||^2
  }
}

// ---------------------------------------------------------------------------
// per-row sum of squares of (x + EPS) over KD elements (8 rows per block)
__global__ __launch_bounds__(256)
void row_sumsq_kernel(const float* __restrict__ X, float* __restrict__ out) {
  int wave = threadIdx.x >> 5, lane = threadIdx.x & 31;
  size_t row = (size_t)blockIdx.x * 8 + wave;
  const float* xp = X + row * KD;
  float s = 0.0f;
  #pragma unroll 4
  for (int i = lane; i < KD; i += 32) { float v = xp[i] + EPSF; s += v * v; }
  #pragma unroll
  for (int off = 16; off > 0; off >>= 1) s += __shfl_xor(s, off);
  if (lane == 0) out[row] = s;
}

// ---------------------------------------------------------------------------
// Raw GEMM on V_WMMA_F32_16X16X4_F32:  Craw[M x 1024] = (A + EPS) * Bm
// MT = number of 16-row M-tiles per workgroup sharing each B fragment.
// Grid: (M/(16*MT)) * (ND/128) blocks, 256 threads (8 waves); wave w owns the
// 16-wide tile column at n0 + 16*w for all MT row tiles.
// Also atomically accumulates per-row sum-of-squares of Craw into sumsq[].
// Optionally zeroes zero_buf (block 0) before downstream accumulation.
template <int MT>
__global__ __launch_bounds__(256)
void mob_gemm_raw_kernel(const float* __restrict__ A,
                         const float* __restrict__ Bm,
                         float* __restrict__ Craw,
                         float* __restrict__ sumsq,
                         float* __restrict__ zero_buf, int zero_n) {
  __shared__ float lA[16 * MT * CHUNK];          // 4 KB (MT=1) / 8 KB (MT=2)
  const int nbn  = ND / 128;                     // 8
  const int mblk = blockIdx.x / nbn;
  const int nblk = blockIdx.x - mblk * nbn;
  const size_t m0 = (size_t)mblk * (16 * MT);
  const int n0   = nblk * 128;
  const int tid  = threadIdx.x;
  const int wave = tid >> 5;
  const int lane = tid & 31;
  const int lhalf = lane >> 4;                   // 0 or 1
  const int l15   = lane & 15;

  if (zero_buf != nullptr && blockIdx.x == 0 && tid < zero_n) zero_buf[tid] = 0.0f;

  v8f acc[MT];
  #pragma unroll
  for (int mt = 0; mt < MT; ++mt)
    acc[mt] = (v8f){0.f, 0.f, 0.f, 0.f, 0.f, 0.f, 0.f, 0.f};

  const int nCol = n0 + wave * 16 + l15;
  const float* Bp = Bm + nCol;

  for (int k0 = 0; k0 < KD; k0 += CHUNK) {
    __syncthreads();                             // protect previous-chunk reads
    // cooperatively stage A[16*MT][CHUNK] (+EPS) into LDS (branchless)
    #pragma unroll
    for (int it = 0; it < (16 * MT * CHUNK) / 256; ++it) {
      int i = tid + it * 256;
      int r = i >> 6, c = i & (CHUNK - 1);
      lA[i] = A[(m0 + r) * KD + (size_t)(k0 + c)] + EPSF;
    }
    __syncthreads();
    #pragma unroll
    for (int kk = 0; kk < CHUNK; kk += 4) {
      // B 4x16 fragment: VGPR0 -> K = kk+2*lhalf, VGPR1 -> K+1, col = nCol.
      // One B fragment feeds all MT row tiles.
      const float* bp = Bp + (size_t)(k0 + kk + 2 * lhalf) * ND;
      v2f bfrag; bfrag.x = bp[0]; bfrag.y = bp[ND];
      #pragma unroll
      for (int mt = 0; mt < MT; ++mt) {
        // A 16x4 fragment: row = l15, K pair base = kk + 2*lhalf (ds b64)
        const float* ap = &lA[(mt * 16 + l15) * CHUNK + kk + 2 * lhalf];
        v2f afrag; afrag.x = ap[0]; afrag.y = ap[1];
        acc[mt] = __builtin_amdgcn_wmma_f32_16x16x4_f32(
            /*neg_a=*/false, afrag, /*neg_b=*/false, bfrag,
            /*c_mod=*/(short)0, acc[mt], /*reuse_a=*/false, /*reuse_b=*/false);
      }
    }
  }

  // Epilogue: store raw C; reduce row sum-of-squares across the 16-lane half.
  #pragma unroll
  for (int mt = 0; mt < MT; ++mt) {
    #pragma unroll
    for (int v = 0; v < 8; ++v) {
      int row = mt * 16 + v + 8 * lhalf;         // C layout: VGPR v, lane half
      float c = acc[mt][v];
      Craw[(m0 + row) * ND + (size_t)nCol] = c;
      float sq = c * c;
      sq += __shfl_xor(sq, 1);
      sq += __shfl_xor(sq, 2);
      sq += __shfl_xor(sq, 4);
      sq += __shfl_xor(sq, 8);
      if (l15 == 0) atomicAdd(&sumsq[m0 + row], sq);
    }
  }
}

// ---------------------------------------------------------------------------
__device__ __forceinline__ void project2(float& x, float& y) {
  float n = sqrtf(x * x + y * y);
  if (n > MAXN) { float s = MAXN / fmaxf(n, EPSF); x *= s; y *= s; }
}

__device__ __forceinline__ void mobadd2(float ux, float uy, float vx, float vy,
                                        float& ox, float& oy) {
  vx += EPSF; vy += EPSF;
  float uv2 = 2.0f * (ux * vx + uy * vy);
  float u2  = ux * ux + uy * uy;
  float v2  = vx * vx + vy * vy;
  float num = 1.0f + uv2 + v2;
  float den = 1.0f + uv2 + u2 * v2;
  float r = 1.0f / den;
  ox = (num * ux + (1.0f - u2) * vx) * r;
  oy = (num * uy + (1.0f - u2) * vy) * r;
  project2(ox, oy);
}

__device__ __forceinline__ float mob_scale(float msumsq, float xsumsq) {
  float xn = fminf(fmaxf(sqrtf(xsumsq), EPSF), MAXN);
  float Mn = fmaxf(sqrtf(msumsq), EPSF);
  return tanhf(Mn / xn * atanhf(xn)) / Mn;
}

// ---------------------------------------------------------------------------
// In-place tanh-norm nonlinearity + 2D projection on Ux (stored in d_out)
__global__ __launch_bounds__(256)
void ux_finish_kernel(float* __restrict__ C, const float* __restrict__ msums,
                      const float* __restrict__ xsums) {
  size_t gid = (size_t)blockIdx.x * 256 + threadIdx.x;   // BB*TT*HH points
  size_t row = gid >> 9;                                 // / HH
  int p = (int)(gid & (HH - 1));
  size_t base = row * ND + 2 * p;
  float s  = mob_scale(msums[row], xsums[row]);
  float rx = C[base] * s, ry = C[base + 1] * s;
  project2(rx, ry);
  C[base] = rx; C[base + 1] = ry;
}

// ---------------------------------------------------------------------------
// h_new = mob_add(mob_add(Wh, Ux), b); writes h state + output; accumulates
// ||h_new+EPS||^2 into hsum[(t+1)%2] and zeroes msum[(t+1)%2].
// Grid: 2*BB blocks x 256 threads (one (b, 256-point) slab per block).
__global__ __launch_bounds__(256)
void combine_kernel(const float* __restrict__ whraw,
                    float* __restrict__ msum, float* __restrict__ hsum,
                    float* __restrict__ out, const float* __restrict__ bias,
                    float* __restrict__ h, int t) {
  __shared__ float red;
  const int b = blockIdx.x >> 1;
  const int p = ((blockIdx.x & 1) << 8) + threadIdx.x;   // 0..HH-1
  if (threadIdx.x == 0) red = 0.0f;
  __syncthreads();
  const int cur = t & 1, nxt = (t + 1) & 1;

  // Wh nonlinearity + projection
  float s  = mob_scale(msum[cur * BB + b], hsum[cur * BB + b]);
  float wx = whraw[b * ND + 2 * p] * s;
  float wy = whraw[b * ND + 2 * p + 1] * s;
  project2(wx, wy);

  // Ux (already projected, stored in out)
  size_t obase = ((size_t)b * TT + t) * ND + 2 * p;
  float uxx = out[obase], uxy = out[obase + 1];

  float c1x, c1y; mobadd2(wx, wy, uxx, uxy, c1x, c1y);
  float hx, hy;   mobadd2(c1x, c1y, bias[2 * p], bias[2 * p + 1], hx, hy);

  out[obase] = hx; out[obase + 1] = hy;
  h[b * ND + 2 * p] = hx; h[b * ND + 2 * p + 1] = hy;

  float e0 = hx + EPSF, e1 = hy + EPSF;
  atomicAdd(&red, e0 * e0 + e1 * e1);            // LDS atomic (ds_add_f32)
  __syncthreads();
  if (threadIdx.x == 0) {
    atomicAdd(&hsum[nxt * BB + b], red);
    if ((blockIdx.x & 1) == 0) msum[nxt * BB + b] = 0.0f;
  }
}

// ---------------------------------------------------------------------------
extern "C" void kernel_launch(void* const* d_in, const int* in_sizes, int n_in,
                              void* d_out, int out_size, void* d_ws, size_t ws_size,
                              hipStream_t stream) {
  const float* inputs = (const float*)d_in[0];   // (B,T,I,2)  == (B*T) x 1024
  const float* w      = (const float*)d_in[1];   // 1024 x 1024 (K x N)
  const float* u      = (const float*)d_in[2];   // 1024 x 1024 (K x N)
  const float* bias   = (const float*)d_in[3];   // (H,2)
  float* out = (float*)d_out;                    // (B,T,H,2)

  // Workspace layout (~2.6 MB total)
  float* h_ws   = (float*)d_ws;                  // BB*1024
  float* wh_raw = h_ws   + BB * KD;              // BB*1024
  float* msum   = wh_raw + BB * KD;              // 2*BB
  float* hsum   = msum   + 2 * BB;               // 2*BB
  float* uxsum  = hsum   + 2 * BB;               // BB*TT
  float* xsum   = uxsum  + BB * TT;              // BB*TT

  // ---- init + per-row ||x+EPS||^2 for every (b,t) row
  rnn_init_kernel<<<(BB * KD) / 256, 256, 0, stream>>>(h_ws, msum, hsum, uxsum);
  row_sumsq_kernel<<<(BB * TT) / 8, 256, 0, stream>>>(inputs, xsum);

  // ---- Ux for all timesteps: one big WMMA GEMM into d_out (raw), then fused
  //      tanh-norm nonlinearity in place.  MT=2: 32 rows/WG, B frag reused.
  mob_gemm_raw_kernel<2><<<(BB * TT / 32) * (ND / 128), 256, 0, stream>>>(
      inputs, u, out, uxsum, nullptr, 0);
  ux_finish_kernel<<<(BB * TT * HH) / 256, 256, 0, stream>>>(out, uxsum, xsum);

  // ---- recurrent chain: per step, Wh GEMM (raw + row norms) then combine.
  //      MT=1 keeps 128 workgroups/step for the latency-bound chain.
  for (int t = 0; t < TT; ++t) {
    mob_gemm_raw_kernel<1><<<(BB / 16) * (ND / 128), 256, 0, stream>>>(
        h_ws, w, wh_raw, msum + (t & 1) * BB,
        /*zero hsum[(t+1)%2] before combine accumulates it*/
        hsum + ((t + 1) & 1) * BB, BB);
    combine_kernel<<<2 * BB, 256, 0, stream>>>(wh_raw, msum, hsum, out, bias,
                                               h_ws, t);
  }
}